// GAT_15753940041793
// MI455X (gfx1250) — compile-verified
//
#include <hip/hip_runtime.h>

#define HEADS 4
#define HID   32
#define F     128   // HEADS*HID
#define NEG_SLOPE 0.2f

typedef __attribute__((ext_vector_type(2))) float v2f;
typedef __attribute__((ext_vector_type(8))) float v8f;

// ---------------------------------------------------------------- utilities
__global__ void fill_f32(float* __restrict__ p, float v, long n) {
    long i = (long)blockIdx.x * blockDim.x + threadIdx.x;
    if (i < n) p[i] = v;
}

// ------------------------------------------------------------ WMMA GEMM
// H[N, 128] = X[N, K] @ W[K, 128]      (K = 128 layer 1, K = 32 layers 2/3)
// grid.x = ceil(N/16); 256 threads = 8 wave32; wave w computes cols [16w,16w+16)
__global__ void gemm_wmma(const float* __restrict__ X, int K,
                          const float* __restrict__ W,
                          float* __restrict__ H, int Nrows) {
    const int lane = threadIdx.x & 31;
    const int wave = threadIdx.x >> 5;
    const int row0 = blockIdx.x * 16;
    const int col0 = wave * 16;
    const int l     = lane & 15;
    const int khalf = (lane >> 4) << 1;        // 0 for lanes 0-15, 2 for 16-31

    int rowA = row0 + l;
    if (rowA >= Nrows) rowA = Nrows - 1;       // clamp (N=100000 is 16-aligned anyway)
    const float* __restrict__ xrow = X + (long)rowA * K;

    v8f c = {0.f, 0.f, 0.f, 0.f, 0.f, 0.f, 0.f, 0.f};
    for (int k = 0; k < K; k += 4) {
        v2f a, b;
        // A fragment: 16x4 tile of X   (M across lanes 0-15, K split by lane half)
        a.x = xrow[k + khalf];
        a.y = xrow[k + khalf + 1];
        // B fragment: 4x16 tile of W   (N across lanes, K split by lane half)
        b.x = W[(k + khalf)     * F + col0 + l];
        b.y = W[(k + khalf + 1) * F + col0 + l];
        c = __builtin_amdgcn_wmma_f32_16x16x4_f32(
                /*neg_a=*/false, a, /*neg_b=*/false, b,
                /*c_mod=*/(short)0, c, /*reuse_a=*/false, /*reuse_b=*/false);
    }

    // C/D layout: VGPR r, lanes 0-15 -> M=r ; lanes 16-31 -> M=8+r ; N = lane&15
    const int nc    = col0 + l;
    const int mbase = row0 + ((lane >> 4) << 3);
#pragma unroll
    for (int r = 0; r < 8; ++r) {
        int m = mbase + r;
        if (m < Nrows) H[(long)m * F + nc] = c[r];
    }
}

// ----------------------------------------------------- per-node attention logits
__global__ void alpha_k(const float* __restrict__ H, long N,
                        const float* __restrict__ a_s, const float* __restrict__ a_d,
                        float* __restrict__ asrc, float* __restrict__ adst) {
    long t = (long)blockIdx.x * blockDim.x + threadIdx.x;
    if (t >= N * HEADS) return;
    long n = t >> 2;
    int  hd = t & 3;
    const float* hr = H + n * F + hd * HID;
    float s = 0.f, d = 0.f;
#pragma unroll
    for (int c = 0; c < HID; ++c) {
        float hv = hr[c];
        s += hv * a_s[hd * HID + c];
        d += hv * a_d[hd * HID + c];
    }
    asrc[t] = s;
    adst[t] = d;
}

__device__ __forceinline__ void edge_sd(const int* __restrict__ ei, long E, long e,
                                        int& s, int& d) {
    if (e < E) { s = ei[e]; d = ei[E + e]; }
    else       { s = d = (int)(e - E); }       // self loops appended
}

__device__ __forceinline__ float lrelu(float v) {
    return v >= 0.f ? v : NEG_SLOPE * v;
}

// ----------------------------------------------------------- segment max (float atomic)
__global__ void edge_max_k(const int* __restrict__ ei, long E, long Etot,
                           const float* __restrict__ asrc, const float* __restrict__ adst,
                           float* __restrict__ emax) {
    long e = (long)blockIdx.x * blockDim.x + threadIdx.x;
    if (e >= Etot) return;
    int s, d; edge_sd(ei, E, e, s, d);
#pragma unroll
    for (int hd = 0; hd < HEADS; ++hd) {
        float v = lrelu(asrc[(long)s * 4 + hd] + adst[(long)d * 4 + hd]);
        float* addr = emax + (long)d * 4 + hd;
        if (v >= 0.f) atomicMax((int*)addr, __float_as_int(v));
        else          atomicMin((unsigned int*)addr, __float_as_uint(v));
    }
}

// ----------------------------------------------------------- exp + segment sum
__global__ void edge_exp_k(const int* __restrict__ ei, long E, long Etot,
                           const float* __restrict__ asrc, const float* __restrict__ adst,
                           const float* __restrict__ emax,
                           float* __restrict__ eexp, float* __restrict__ denom) {
    long e = (long)blockIdx.x * blockDim.x + threadIdx.x;
    if (e >= Etot) return;
    int s, d; edge_sd(ei, E, e, s, d);
#pragma unroll
    for (int hd = 0; hd < HEADS; ++hd) {
        float v  = lrelu(asrc[(long)s * 4 + hd] + adst[(long)d * 4 + hd]);
        float ex = __expf(v - emax[(long)d * 4 + hd]);
        eexp[e * 4 + hd] = ex;
        atomicAdd(denom + (long)d * 4 + hd, ex);
    }
}

// ----------------------------------------------------------- weighted scatter-add
// one thread per (edge, channel-of-32); loops over 4 heads
__global__ void edge_aggr_k(const int* __restrict__ ei, long E, long Etot,
                            const float* __restrict__ H,
                            const float* __restrict__ eexp, const float* __restrict__ denom,
                            float* __restrict__ aggr) {
    long gid = (long)blockIdx.x * blockDim.x + threadIdx.x;
    long e = gid >> 5;
    int  cc = (int)(gid & 31);
    if (e >= Etot) return;
    int s, d; edge_sd(ei, E, e, s, d);
    const float* hs = H + (long)s * F;
    __builtin_prefetch(hs + cc, 0, 1);         // global_prefetch_b8 on gathered row
#pragma unroll
    for (int hd = 0; hd < HEADS; ++hd) {
        float coef = eexp[e * 4 + hd] / (denom[(long)d * 4 + hd] + 1e-16f);
        atomicAdd(aggr + (long)d * F + hd * HID + cc, hs[hd * HID + cc] * coef);
    }
}

// ----------------------------------------------------------- head mean + bias + relu
__global__ void finalize_k(const float* __restrict__ aggr, const float* __restrict__ b,
                           long N, float* __restrict__ xout) {
    long t = (long)blockIdx.x * blockDim.x + threadIdx.x;
    if (t >= N * HID) return;
    long n = t >> 5;
    int  c = (int)(t & 31);
    const float* ar = aggr + n * F;
    float v = 0.25f * (ar[c] + ar[HID + c] + ar[2 * HID + c] + ar[3 * HID + c]) + b[c];
    xout[t] = v > 0.f ? v : 0.f;
}

// ----------------------------------------------------------- global mean pool (stage 1)
__global__ void node_reduce_k(const float* __restrict__ xb, long N, float* __restrict__ gsum) {
    long t = (long)blockIdx.x * blockDim.x + threadIdx.x;   // 32768 threads total
    int  c  = (int)(t & 31);
    long n0 = t >> 5;                                        // 1024 strides per channel
    float acc = 0.f;
    for (long n = n0; n < N; n += 1024) acc += xb[n * HID + c];
    atomicAdd(gsum + c, acc);
}

// ----------------------------------------------------------- MLP head (1 thread)
__global__ void head_k(const float* __restrict__ gsum, long N,
                       const float* __restrict__ lw1, const float* __restrict__ lb1,
                       const float* __restrict__ lw2, const float* __restrict__ lb2,
                       float* __restrict__ out) {
    if (threadIdx.x != 0 || blockIdx.x != 0) return;
    float g[HID];
#pragma unroll
    for (int c = 0; c < HID; ++c) g[c] = gsum[c] / (float)N;
    float acc = lb2[0];
#pragma unroll
    for (int j = 0; j < HID / 2; ++j) {
        float tt = lb1[j];
#pragma unroll
        for (int c = 0; c < HID; ++c) tt += g[c] * lw1[c * (HID / 2) + j];
        tt = tt > 0.f ? tt : 0.f;
        acc += tt * lw2[j];
    }
    out[0] = acc;
}

// ================================================================ launcher
static inline int nblk(long n, int t) { return (int)((n + t - 1) / t); }

extern "C" void kernel_launch(void* const* d_in, const int* in_sizes, int n_in,
                              void* d_out, int out_size, void* d_ws, size_t ws_size,
                              hipStream_t stream) {
    (void)n_in; (void)out_size; (void)ws_size;
    const float* x  = (const float*)d_in[0];
    const int*   ei = (const int*)d_in[1];
    const float* Wl[3]  = {(const float*)d_in[2],  (const float*)d_in[6],  (const float*)d_in[10]};
    const float* asl[3] = {(const float*)d_in[3],  (const float*)d_in[7],  (const float*)d_in[11]};
    const float* adl[3] = {(const float*)d_in[4],  (const float*)d_in[8],  (const float*)d_in[12]};
    const float* bl[3]  = {(const float*)d_in[5],  (const float*)d_in[9],  (const float*)d_in[13]};
    const float* lw1 = (const float*)d_in[14];
    const float* lb1 = (const float*)d_in[15];
    const float* lw2 = (const float*)d_in[16];
    const float* lb2 = (const float*)d_in[17];

    const long N    = in_sizes[0] / F;        // 100000
    const long E    = in_sizes[1] / 2;        // 1600000
    const long Etot = E + N;

    // ------- workspace carve-up (floats)
    float* ws    = (float*)d_ws;
    float* hbuf  = ws;                         // N*F
    float* aggr  = hbuf  + N * F;              // N*F
    float* xbuf  = aggr  + N * F;              // N*HID
    float* asrc  = xbuf  + N * HID;            // N*4
    float* adst  = asrc  + N * 4;              // N*4
    float* emax  = adst  + N * 4;              // N*4
    float* denom = emax  + N * 4;              // N*4
    float* eexp  = denom + N * 4;              // Etot*4
    float* gsum  = eexp  + Etot * 4;           // 32

    const int T = 256;
    const float NEG_INF = __builtin_huge_valf() * -1.f;

    for (int layer = 0; layer < 3; ++layer) {
        const int   K  = (layer == 0) ? F : HID;
        const float* X = (layer == 0) ? x : xbuf;

        fill_f32<<<nblk(N * 4, T), T, 0, stream>>>(emax,  NEG_INF, N * 4);
        fill_f32<<<nblk(N * 4, T), T, 0, stream>>>(denom, 0.f,     N * 4);
        fill_f32<<<nblk(N * F, T), T, 0, stream>>>(aggr,  0.f,     N * F);

        gemm_wmma<<<nblk(N, 16), 256, 0, stream>>>(X, K, Wl[layer], hbuf, (int)N);
        alpha_k<<<nblk(N * 4, T), T, 0, stream>>>(hbuf, N, asl[layer], adl[layer], asrc, adst);
        edge_max_k<<<nblk(Etot, T), T, 0, stream>>>(ei, E, Etot, asrc, adst, emax);
        edge_exp_k<<<nblk(Etot, T), T, 0, stream>>>(ei, E, Etot, asrc, adst, emax, eexp, denom);
        edge_aggr_k<<<nblk(Etot * 32, T), T, 0, stream>>>(ei, E, Etot, hbuf, eexp, denom, aggr);
        finalize_k<<<nblk(N * HID, T), T, 0, stream>>>(aggr, bl[layer], N, xbuf);
    }

    fill_f32<<<1, 32, 0, stream>>>(gsum, 0.f, 32);
    node_reduce_k<<<128, 256, 0, stream>>>(xbuf, N, gsum);
    head_k<<<1, 32, 0, stream>>>(gsum, N, lw1, lb1, lw2, lb2, (float*)d_out);
}